// VLAQ_26645977104735
// MI455X (gfx1250) — compile-verified
//
#include <hip/hip_runtime.h>
#include <cstddef>
#include <cstdint>

// CDNA5 / gfx1250, wave32. Fused single-pass VLAD aggregation.
// Bandwidth-bound: one 51.2 MB streaming read of tokens (~2.2us at 23.3 TB/s).
// Per 16-token chunk: scores^T = (q/8) @ z^T via 16 V_WMMA_F32_16X16X4_F32,
// e = exp(scores) (no max shift: alpha = e/denom is shift-invariant and the
// N(0,1)-scale scores cannot overflow f32 exp), then acc += e^T @ z via 16
// more WMMAs. Tokens are staged global->LDS with GLOBAL_LOAD_ASYNC_TO_LDS_B128
// (ASYNCcnt-tracked, zero VGPR staging) into a double buffer per wave.
// 4 waves per segment, partial sums combined through LDS (no max-rescale
// needed, so partials combine by plain addition).

typedef __attribute__((ext_vector_type(2))) float v2f;
typedef __attribute__((ext_vector_type(8))) float v8f;

#define S_Q      16
#define C_DIM    64
#define ZS_STR   68   // padded: bank = (68r + c) % 64 = (4r + c) % 64 -> conflict-free
#define ES_STR   20   // padded: 20r % 64 distinct for r = 0..15
#define NW       4    // waves per segment
#define F32_EPS  1.1920929e-07f
#define NORM_EPS 1e-12f

// Compiler + HW ordering point for intra-wave LDS producer->consumer.
#define WAVE_LDS_FENCE() asm volatile("s_wait_dscnt 0x0" ::: "memory")

// 16 tokens x 64 ch: 8 x B128 per wave (32 lanes x 16B each = 512B/instr).
// inst offset applies to BOTH the global and LDS address -> contiguous spans.
__device__ __forceinline__ void async_load_tile(uint32_t lds_addr, uint64_t gbase,
                                                uint32_t goff) {
    asm volatile(
        "global_load_async_to_lds_b128 %0, %1, %2 offset:0\n\t"
        "global_load_async_to_lds_b128 %0, %1, %2 offset:16\n\t"
        "global_load_async_to_lds_b128 %0, %1, %2 offset:32\n\t"
        "global_load_async_to_lds_b128 %0, %1, %2 offset:48\n\t"
        "global_load_async_to_lds_b128 %0, %1, %2 offset:64\n\t"
        "global_load_async_to_lds_b128 %0, %1, %2 offset:80\n\t"
        "global_load_async_to_lds_b128 %0, %1, %2 offset:96\n\t"
        "global_load_async_to_lds_b128 %0, %1, %2 offset:112"
        :: "v"(lds_addr), "v"(goff), "s"(gbase) : "memory");
}

__device__ __forceinline__ float half_sum16(float v) {
    v += __shfl_xor(v, 1, 32);
    v += __shfl_xor(v, 2, 32);
    v += __shfl_xor(v, 4, 32);
    v += __shfl_xor(v, 8, 32);
    return v;
}
__device__ __forceinline__ float wave_sum32(float v) {
    v += __shfl_xor(v, 16, 32);
    v += __shfl_xor(v, 8, 32);
    v += __shfl_xor(v, 4, 32);
    v += __shfl_xor(v, 2, 32);
    v += __shfl_xor(v, 1, 32);
    return v;
}

// Prologue: lower_bound of each batch id in the sorted batch_indices array.
__global__ void vlaq_seg_bounds(const int* __restrict__ bidx, int T, int B,
                                int* __restrict__ seg) {
    int b = blockIdx.x * blockDim.x + threadIdx.x;
    if (b > B) return;
    int lo = 0, hi = T;
    while (lo < hi) {
        int mid = (lo + hi) >> 1;
        if (bidx[mid] < b) lo = mid + 1; else hi = mid;
    }
    seg[b] = lo;   // seg[B] naturally == T
}

__launch_bounds__(32 * NW, 1)
__global__ void vlaq_aggregate(const float* __restrict__ tokens,
                               const float* __restrict__ qk,
                               const int* __restrict__ seg,
                               float* __restrict__ out) {
    __shared__ float zs[NW][2][16 * ZS_STR];  // per-wave double-buffered z tile
    __shared__ float es[NW][16 * ES_STR];     // per-wave exp tile [s][t]
    __shared__ float qs[S_Q * ZS_STR];        // queries [s][c] (unscaled)
    __shared__ float comb[(NW - 1) * 32 * 32];// cross-wave acc combine
    __shared__ float dnc[(NW - 1) * 32 * 8];  // cross-wave denom combine

    const int tid  = threadIdx.x;
    const int w    = tid >> 5;
    const int lane = tid & 31;
    const bool hi  = lane >= 16;
    const int l16  = lane & 15;
    const int koff = hi ? 2 : 0;        // K-offset of VGPR0 for f32 A/B operands

    const int b  = blockIdx.x;
    const int t0 = seg[b];
    const int t1 = seg[b + 1];

    // ---- stage q into LDS (one wave, coalesced float4) ----
    const int row  = lane >> 1;          // 2 lanes per token row
    const int colb = (lane & 1) * 32;    // 32 floats (= 8 x 16B) per lane
    if (tid < 32) {
        const float4* src = (const float4*)(qk + (size_t)row * C_DIM + colb);
        #pragma unroll
        for (int j = 0; j < 8; ++j)
            *(float4*)&qs[row * ZS_STR + colb + 4 * j] = src[j];
    }
    __syncthreads();

    // ---- q in WMMA-A layout (M = s = l16, K = channel), pre-scaled by 1/sqrt(C) ----
    v2f aq[16];
    #pragma unroll
    for (int kk = 0; kk < 16; ++kk) {
        aq[kk].x = qs[l16 * ZS_STR + 4 * kk + koff]     * 0.125f;
        aq[kk].y = qs[l16 * ZS_STR + 4 * kk + koff + 1] * 0.125f;
    }

    // ---- per-wave accumulators ----
    float denp[8];                 // lane-local partial denominators
    v8f acc[4];                    // acc[ct]: rows s = r + hi*8, cols ct*16 + l16
    #pragma unroll
    for (int r = 0; r < 8; ++r) denp[r] = 0.0f;
    #pragma unroll
    for (int ct = 0; ct < 4; ++ct) acc[ct] = (v8f)0.0f;

    // LDS byte addresses for this lane's async-copy slice (both buffers)
    const uint32_t zs_base = (uint32_t)(uintptr_t)(&zs[w][0][0]);
    const uint32_t lds_lane = (uint32_t)(row * ZS_STR + colb) * 4u;
    const uint32_t lds_a0 = zs_base + lds_lane;
    const uint32_t lds_a1 = zs_base + (uint32_t)(16 * ZS_STR * 4) + lds_lane;
    const uint64_t gbase = (uint64_t)(uintptr_t)tokens;

    const int tstart = t0 + w * 16;
    int buf = 0;

    // ---- kick off this wave's first chunk (tail rows clamped to t1-1) ----
    if (tstart < t1) {
        const int rt = min(tstart + row, t1 - 1);
        async_load_tile(lds_a0, gbase, (uint32_t)rt * (C_DIM * 4) + colb * 4);
    }

    for (int t = tstart; t < t1; t += 16 * NW) {
        const int nt = min(16, t1 - t);
        const uint32_t zoff = buf ? (uint32_t)(16 * ZS_STR) : 0u;
        const float* zsb = &zs[w][0][0] + zoff;

        // software pipeline: launch next chunk into the other buffer, then
        // wait for the current chunk (8 newer async ops may stay in flight).
        const int tn = t + 16 * NW;
        if (tn < t1) {
            const int rt = min(tn + row, t1 - 1);
            async_load_tile(buf ? lds_a0 : lds_a1, gbase,
                            (uint32_t)rt * (C_DIM * 4) + colb * 4);
            asm volatile("s_wait_asynccnt 0x8" ::: "memory");
        } else {
            asm volatile("s_wait_asynccnt 0x0" ::: "memory");
        }

        // ---- scores^T = (q/8) @ z^T : 16 K=4 WMMAs, two interleaved chains ----
        v8f s0 = (v8f)0.0f, s1 = (v8f)0.0f;
        #pragma unroll
        for (int kk2 = 0; kk2 < 8; ++kk2) {
            v2f bz0, bz1;
            bz0.x = zsb[l16 * ZS_STR + 8 * kk2 + koff];
            bz0.y = zsb[l16 * ZS_STR + 8 * kk2 + koff + 1];
            bz1.x = zsb[l16 * ZS_STR + 8 * kk2 + 4 + koff];
            bz1.y = zsb[l16 * ZS_STR + 8 * kk2 + 4 + koff + 1];
            s0 = __builtin_amdgcn_wmma_f32_16x16x4_f32(
                     false, aq[2 * kk2],     false, bz0, (short)0, s0, false, false);
            s1 = __builtin_amdgcn_wmma_f32_16x16x4_f32(
                     false, aq[2 * kk2 + 1], false, bz1, (short)0, s1, false, false);
        }
        const v8f sT = s0 + s1;

        // ---- e = exp(score), lane-local denom partials, relayout via LDS ----
        const bool valid = l16 < nt;
        #pragma unroll
        for (int r = 0; r < 8; ++r) {
            float ev = valid ? __expf(sT[r]) : 0.0f;
            denp[r] += ev;
            es[w][(r + (hi ? 8 : 0)) * ES_STR + l16] = ev;
        }
        WAVE_LDS_FENCE();

        // ---- acc[s,c] += e^T @ z : 4 independent column-tile chains ----
        #pragma unroll
        for (int ct = 0; ct < 4; ++ct) {
            v8f c = acc[ct];
            #pragma unroll
            for (int kk = 0; kk < 4; ++kk) {
                v2f ae, bz;
                ae.x = es[w][l16 * ES_STR + 4 * kk + koff];
                ae.y = es[w][l16 * ES_STR + 4 * kk + koff + 1];
                bz.x = zsb[(4 * kk + koff)     * ZS_STR + ct * 16 + l16];
                bz.y = zsb[(4 * kk + koff + 1) * ZS_STR + ct * 16 + l16];
                c = __builtin_amdgcn_wmma_f32_16x16x4_f32(
                        false, ae, false, bz, (short)0, c, false, false);
            }
            acc[ct] = c;
        }
        WAVE_LDS_FENCE();   // drain DS reads before async engine refills buffer
        buf ^= 1;
    }

    // ---- combine the NW waves' partial sums (plain addition) ----
    __syncthreads();
    if (w > 0) {
        float* cb = &comb[(w - 1) * 32 * 32 + lane * 32];
        float* db = &dnc[(w - 1) * 32 * 8 + lane * 8];
        #pragma unroll
        for (int ct = 0; ct < 4; ++ct)
            #pragma unroll
            for (int r = 0; r < 8; ++r) cb[ct * 8 + r] = acc[ct][r];
        #pragma unroll
        for (int r = 0; r < 8; ++r) db[r] = denp[r];
    }
    __syncthreads();
    if (w != 0) return;

    #pragma unroll
    for (int ww = 0; ww < NW - 1; ++ww) {
        const float* cb = &comb[ww * 32 * 32 + lane * 32];
        const float* db = &dnc[ww * 32 * 8 + lane * 8];
        #pragma unroll
        for (int ct = 0; ct < 4; ++ct)
            #pragma unroll
            for (int r = 0; r < 8; ++r) acc[ct][r] += cb[ct * 8 + r];
        #pragma unroll
        for (int r = 0; r < 8; ++r) denp[r] += db[r];
    }

    // ---- epilogue: residual, intra-row L2, global L2, store (wave 0) ----
    float tot = 0.0f;
    #pragma unroll
    for (int r = 0; r < 8; ++r) {
        const float denom = half_sum16(denp[r]);
        const float dc = fmaxf(denom, F32_EPS);
        const float sw = denom / dc;
        const int s = r + (hi ? 8 : 0);
        float res[4], ss = 0.0f;
        #pragma unroll
        for (int ct = 0; ct < 4; ++ct) {
            float v = acc[ct][r] / dc - qs[s * ZS_STR + ct * 16 + l16] * sw;
            res[ct] = v;
            ss += v * v;
        }
        ss = half_sum16(ss);
        const float inv = 1.0f / fmaxf(sqrtf(ss), NORM_EPS);
        #pragma unroll
        for (int ct = 0; ct < 4; ++ct) {
            float v = res[ct] * inv;
            acc[ct][r] = v;
            tot += v * v;
        }
    }
    tot = wave_sum32(tot);
    const float inv2 = 1.0f / fmaxf(sqrtf(tot), NORM_EPS);

    float* ob = out + (size_t)b * (S_Q * C_DIM);
    #pragma unroll
    for (int r = 0; r < 8; ++r) {
        const int s = r + (hi ? 8 : 0);
        #pragma unroll
        for (int ct = 0; ct < 4; ++ct)
            ob[s * C_DIM + ct * 16 + l16] = acc[ct][r] * inv2;
    }
}

extern "C" void kernel_launch(void* const* d_in, const int* in_sizes, int n_in,
                              void* d_out, int out_size, void* d_ws, size_t ws_size,
                              hipStream_t stream) {
    const float* tokens = (const float*)d_in[0];
    const float* qk     = (const float*)d_in[1];
    const int*   bidx   = (const int*)d_in[2];

    const int T = in_sizes[0] / C_DIM;
    const int B = out_size / (S_Q * C_DIM);

    int* seg = (int*)d_ws;   // B+1 ints of scratch

    vlaq_seg_bounds<<<(B + 1 + 255) / 256, 256, 0, stream>>>(bidx, T, B, seg);
    vlaq_aggregate<<<B, 32 * NW, 0, stream>>>(tokens, qk, seg, (float*)d_out);
}